// S4DKernel_86122684219544
// MI455X (gfx1250) — compile-verified
//
#include <hip/hip_runtime.h>

typedef __attribute__((ext_vector_type(2))) float v2f;
typedef __attribute__((ext_vector_type(4))) float v4f;
typedef __attribute__((ext_vector_type(8))) float v8f;

#define S4D_D 1024
#define S4D_N 64
#define S4D_L 2048

// One wave (32 lanes, wave32) per channel d.
// K[d, 16*j + k] = Re( sum_n w[n] * Abar[n]^(16j+k) )
//               = ( P @ Q )[k, j]   with the real-part trick:
//   P_re[k][n] =  Re(w * Abar^k)      (A matrix, 16 x 64)
//   P_nim[k][n]= -Im(w * Abar^k)
//   Q_re[j][n] =  Re((Abar^16)^j)     (B matrix stored j-major, 64 x 16 tile)
//   Q_im[j][n] =  Im((Abar^16)^j)
// acc = P_re @ Q_re + P_nim @ Q_im  via V_WMMA_F32_16X16X4_F32 (K-dim = n),
// kept as two independent WMMA accumulator chains, merged at the end.
__global__ __launch_bounds__(32) void s4d_wmma_kernel(
    const float* __restrict__ A_real,   // (64,)
    const float* __restrict__ A_imag,   // (64,)
    const float* __restrict__ Bmat,     // (64, 1024)
    const float* __restrict__ Cmat,     // (1024, 64, 2)
    const float* __restrict__ Dvec,     // (1024,)
    const float* __restrict__ log_dt,   // (1024,)
    float* __restrict__ out)            // (1024, 2048)
{
    __shared__ float P_re [16 * 64];
    __shared__ float P_nim[16 * 64];
    __shared__ float Q_re [16 * 64];
    __shared__ float Q_im [16 * 64];

    const int d    = blockIdx.x;
    const int lane = threadIdx.x;     // 0..31
    const int hi   = lane >> 4;       // 0 | 1
    const int lo   = lane & 15;

    const float dt = expf(log_dt[d]);
    const float Dd = Dvec[d];

    // ---- per-lane: 2 owned states n = 2*lane, 2*lane+1 -------------------
    float ar[2], ai[2];   // Abar
    float pr[2], pi[2];   // running w * Abar^k
    #pragma unroll
    for (int t = 0; t < 2; ++t) {
        const int n = 2 * lane + t;
        const float zr = 0.5f * dt * A_real[n];          // dt*A/2
        const float zi = 0.5f * dt * A_imag[n];
        const float dr = 1.0f - zr, di = -zi;            // 1 - dtA/2
        const float inv = 1.0f / (dr * dr + di * di);
        const float nr = 1.0f + zr, ni = zi;             // 1 + dtA/2
        ar[t] = (nr * dr + ni * di) * inv;               // bilinear Abar
        ai[t] = (ni * dr - nr * di) * inv;
        // B_bar = dt * B[n,d] / (1 - dtA/2)
        const float bs  = dt * Bmat[n * S4D_D + d] * inv;
        const float bbr =  bs * dr;
        const float bbi = -bs * di;
        // w = Cc * B_bar
        const float cr = Cmat[(d * S4D_N + n) * 2 + 0];
        const float ci = Cmat[(d * S4D_N + n) * 2 + 1];
        pr[t] = cr * bbr - ci * bbi;
        pi[t] = cr * bbi + ci * bbr;
    }

    // ---- build P into LDS (16 k-steps of the complex recurrence) ---------
    #pragma unroll
    for (int k = 0; k < 16; ++k) {
        v2f sre, sim;
        sre[0] = pr[0];  sre[1] = pr[1];
        sim[0] = -pi[0]; sim[1] = -pi[1];
        *(v2f*)&P_re [k * 64 + 2 * lane] = sre;
        *(v2f*)&P_nim[k * 64 + 2 * lane] = sim;
        #pragma unroll
        for (int t = 0; t < 2; ++t) {
            const float trr = pr[t] * ar[t] - pi[t] * ai[t];
            pi[t] = pr[t] * ai[t] + pi[t] * ar[t];
            pr[t] = trr;
        }
    }

    // ---- Abar^16 by 4 complex squarings ----------------------------------
    float a16r[2], a16i[2];
    #pragma unroll
    for (int t = 0; t < 2; ++t) { a16r[t] = ar[t]; a16i[t] = ai[t]; }
    #pragma unroll
    for (int s = 0; s < 4; ++s) {
        #pragma unroll
        for (int t = 0; t < 2; ++t) {
            const float trr = a16r[t] * a16r[t] - a16i[t] * a16i[t];
            a16i[t] = 2.0f * a16r[t] * a16i[t];
            a16r[t] = trr;
        }
    }

    // running (Abar^16)^j across all 8 column tiles (j = 0..127)
    float qrr[2] = {1.0f, 1.0f}, qii[2] = {0.0f, 0.0f};

    for (int jt = 0; jt < 8; ++jt) {
        __syncthreads();
        // ---- build Q tile (16 columns) into LDS --------------------------
        #pragma unroll
        for (int jl = 0; jl < 16; ++jl) {
            v2f sre, sim;
            sre[0] = qrr[0]; sre[1] = qrr[1];
            sim[0] = qii[0]; sim[1] = qii[1];
            *(v2f*)&Q_re[jl * 64 + 2 * lane] = sre;
            *(v2f*)&Q_im[jl * 64 + 2 * lane] = sim;
            #pragma unroll
            for (int t = 0; t < 2; ++t) {
                const float trr = qrr[t] * a16r[t] - qii[t] * a16i[t];
                qii[t] = qrr[t] * a16i[t] + qii[t] * a16r[t];
                qrr[t] = trr;
            }
        }
        __syncthreads();

        // ---- 16x16 output tile: 16 K-steps x 2 independent WMMA chains ---
        // A-frag (f32 16x4): lane row M = lo; lanes<16 hold K 0..1, lanes>=16 hold K 2..3
        // B-frag (f32 4x16): lane col N = lo; same K-row split -> identical LDS addressing
        v8f acc_r = {};
        v8f acc_i = {};
        #pragma unroll
        for (int kk = 0; kk < 16; ++kk) {
            const int nb = 4 * kk + 2 * hi;
            v2f a_re = *(const v2f*)&P_re [lo * 64 + nb];
            v2f a_im = *(const v2f*)&P_nim[lo * 64 + nb];
            v2f b_re = *(const v2f*)&Q_re [lo * 64 + nb];
            v2f b_im = *(const v2f*)&Q_im [lo * 64 + nb];
            acc_r = __builtin_amdgcn_wmma_f32_16x16x4_f32(
                false, a_re, false, b_re, (short)0, acc_r, false, false);
            acc_i = __builtin_amdgcn_wmma_f32_16x16x4_f32(
                false, a_im, false, b_im, (short)0, acc_i, false, false);
        }

        // ---- store: C/D layout -> row M = v + 8*hi, col j = jt*16 + lo ---
        // l = 16*j + M  => each half-lane writes 8 contiguous floats
        // Merge the two chains, add D at l==0, pack into two b128 stores.
        const int lbase = (jt * 16 + lo) * 16 + 8 * hi;
        v4f w0, w1;
        #pragma unroll
        for (int v = 0; v < 4; ++v) {
            float val = acc_r[v] + acc_i[v];
            val += (lbase + v == 0) ? Dd : 0.0f;   // K[:,0] += D, branch-free
            w0[v] = val;
        }
        #pragma unroll
        for (int v = 0; v < 4; ++v) {
            w1[v] = acc_r[v + 4] + acc_i[v + 4];
        }
        float* orow = out + (size_t)d * S4D_L + lbase;
        *(v4f*)(orow)     = w0;
        *(v4f*)(orow + 4) = w1;
    }
}

extern "C" void kernel_launch(void* const* d_in, const int* in_sizes, int n_in,
                              void* d_out, int out_size, void* d_ws, size_t ws_size,
                              hipStream_t stream) {
    const float* A_real = (const float*)d_in[0];
    const float* A_imag = (const float*)d_in[1];
    const float* B      = (const float*)d_in[2];
    const float* C      = (const float*)d_in[3];
    const float* D      = (const float*)d_in[4];
    const float* log_dt = (const float*)d_in[5];
    // d_in[6] = L (python int scalar); layout specialized for L = 2048
    (void)in_sizes; (void)n_in; (void)out_size; (void)d_ws; (void)ws_size;

    s4d_wmma_kernel<<<S4D_D, 32, 0, stream>>>(A_real, A_imag, B, C, D, log_dt,
                                              (float*)d_out);
}